// GatLayer_33990371180844
// MI455X (gfx1250) — compile-verified
//
#include <hip/hip_runtime.h>
#include <hip/hip_bf16.h>

// GAT layer for MI455X (gfx1250, wave32).
// Phase 1: wx = X @ W via V_WMMA_F32_16X16X4_F32 (fp32-exact matrix pipe).
//          B operand staged in LDS *transposed + padded* so each fragment is
//          one conflict-free ds_load_b64 into an even-aligned VGPR pair.
// Phase 2: per-node attention scores.
// Phase 3-5: edge scatter-softmax + aggregation (HBM/atomic bound).

#define D_IN   128
#define D_OUT  64
#define LEAKY  0.2f

typedef float v2f __attribute__((ext_vector_type(2)));
typedef float v8f __attribute__((ext_vector_type(8)));

// ---- ordered-uint encoding of float: monotone map so atomicMax(u32) == float max
__device__ __forceinline__ unsigned ord_of_float(float f) {
    unsigned u = __float_as_uint(f);
    return (u & 0x80000000u) ? ~u : (u | 0x80000000u);
}
__device__ __forceinline__ float float_of_ord(unsigned u) {
    return __uint_as_float((u & 0x80000000u) ? (u ^ 0x80000000u) : ~u);
}

// ------------------------------------------------------------------ init
__global__ __launch_bounds__(256) void k_init(float* __restrict__ out,
                                              unsigned* __restrict__ segmax,
                                              float* __restrict__ denom,
                                              int n_nodes) {
    int i = blockIdx.x * blockDim.x + threadIdx.x;
    int total = n_nodes * D_OUT;
    if (i < total) out[i] = 0.0f;
    if (i < n_nodes) { segmax[i] = 0u; denom[i] = 0.0f; }
}

// ------------------------------------------------------------------ GEMM
// One wave computes a 16x64 tile of wx. A-frag (16x4 f32): lane L holds
// row = L%16, k-pair (L<16 ? k0,k1 : k2,k3). B-frag (4x16 f32): lane L holds
// col = L%16, same k-pair split. C/D (16x16 f32, 8 VGPRs): reg r -> row
// r + (L>=16 ? 8 : 0), col = L%16.
//
// LDS holds W transposed with padded leading dim LDW=132 so a B fragment is a
// contiguous 8B-aligned v2f: addr/4 = col*132 + k -> bank = (4*(col%16)+k)%64,
// conflict-free across the 32 lanes of a ds_load_b64.
__global__ __launch_bounds__(256) void k_gemm(const float* __restrict__ X,
                                              const float* __restrict__ W,
                                              float* __restrict__ wx,
                                              int n_nodes) {
    constexpr int LDW = D_IN + 4;          // 132 floats; 528B rows (8B aligned)
    __shared__ float sWt[D_OUT * LDW];     // ~33.8 KB of the 320 KB WGP pool
    for (int i = threadIdx.x; i < D_IN * D_OUT; i += blockDim.x) {
        int k = i / D_OUT, col = i % D_OUT;   // W is row-major [k][col]
        sWt[col * LDW + k] = W[i];
    }
    __syncthreads();

    const int lane = threadIdx.x & 31;
    const int wave = threadIdx.x >> 5;
    const int tile = blockIdx.x * (blockDim.x >> 5) + wave;  // 16-row tile id
    const int row0 = tile * 16;
    if (row0 >= n_nodes) return;  // uniform per wave -> EXEC stays all-ones

    const int m    = lane & 15;
    const int koff = (lane >> 4) * 2;          // 0 or 2
    int arow = row0 + m;
    if (arow >= n_nodes) arow = n_nodes - 1;   // clamp (stores guarded below)

    v8f acc[4] = {v8f{}, v8f{}, v8f{}, v8f{}};

    const float* __restrict__ xrow = X + (size_t)arow * D_IN;
    #pragma unroll
    for (int kb = 0; kb < D_IN; kb += 4) {
        v2f a = *(const v2f*)(xrow + kb + koff);   // global_load_b64
        #pragma unroll
        for (int t = 0; t < 4; ++t) {
            const int col = t * 16 + m;
            v2f b = *(const v2f*)(&sWt[col * LDW + kb + koff]);  // ds_load_b64
            acc[t] = __builtin_amdgcn_wmma_f32_16x16x4_f32(
                /*neg_a=*/false, a, /*neg_b=*/false, b,
                /*c_mod=*/(short)0, acc[t],
                /*reuse_a=*/false, /*reuse_b=*/false);
        }
    }

    const int rbase = row0 + ((lane >> 4) * 8);
    #pragma unroll
    for (int t = 0; t < 4; ++t) {
        #pragma unroll
        for (int r = 0; r < 8; ++r) {
            const int row = rbase + r;
            if (row < n_nodes)
                wx[(size_t)row * D_OUT + t * 16 + m] = acc[t][r];
        }
    }
}

// --------------------------------------------------------------- scores
__global__ __launch_bounds__(256) void k_scores(const float* __restrict__ wx,
                                                const float* __restrict__ att,
                                                float* __restrict__ s_src,
                                                float* __restrict__ s_trg,
                                                int n_nodes) {
    int node = blockIdx.x * blockDim.x + threadIdx.x;
    if (node >= n_nodes) return;
    const float* row = wx + (size_t)node * D_OUT;
    float ss = 0.0f, st = 0.0f;
    #pragma unroll 8
    for (int d = 0; d < D_OUT; ++d) {
        float v = row[d];
        ss = fmaf(v, att[d], ss);
        st = fmaf(v, att[D_OUT + d], st);
    }
    s_src[node] = ss;
    s_trg[node] = st;
}

// ------------------------------------------------- edge logits + seg max
__global__ __launch_bounds__(256) void k_logits(const int* __restrict__ esrc,
                                                const int* __restrict__ etrg,
                                                const float* __restrict__ s_src,
                                                const float* __restrict__ s_trg,
                                                float* __restrict__ logits,
                                                unsigned* __restrict__ segmax,
                                                int n_edges) {
    int e = blockIdx.x * blockDim.x + threadIdx.x;
    if (e >= n_edges) return;
    int s = esrc[e], t = etrg[e];
    float l = s_src[s] + s_trg[t];
    l = (l >= 0.0f) ? l : LEAKY * l;
    logits[e] = l;
    atomicMax(&segmax[s], ord_of_float(l));  // global_atomic_max_u32
}

// ---------------------------------------------------- exp + seg sum
__global__ __launch_bounds__(256) void k_exp(const int* __restrict__ esrc,
                                             float* __restrict__ logits,  // in: logit, out: exp
                                             const unsigned* __restrict__ segmax,
                                             float* __restrict__ denom,
                                             int n_edges) {
    int e = blockIdx.x * blockDim.x + threadIdx.x;
    if (e >= n_edges) return;
    int s = esrc[e];
    float ex = __expf(logits[e] - float_of_ord(segmax[s]));
    logits[e] = ex;
    atomicAdd(&denom[s], ex);                // global_atomic_add_f32
}

// ------------------------------------------- aggregation: warp per edge
__global__ __launch_bounds__(256) void k_scatter(const int* __restrict__ esrc,
                                                 const int* __restrict__ etrg,
                                                 const float* __restrict__ ex,
                                                 const float* __restrict__ denom,
                                                 const float* __restrict__ wx,
                                                 float* __restrict__ out,
                                                 int n_edges) {
    int e    = (blockIdx.x * blockDim.x + threadIdx.x) >> 5;  // wave32: edge per warp
    int lane = threadIdx.x & 31;
    if (e >= n_edges) return;
    int s = esrc[e], t = etrg[e];
    float alpha = ex[e] / denom[s];
    const float2* src2 = (const float2*)(wx + (size_t)t * D_OUT);
    float2 v = src2[lane];                                    // dims 2*lane, 2*lane+1
    float* o = out + (size_t)s * D_OUT + 2 * lane;
    atomicAdd(o,     alpha * v.x);
    atomicAdd(o + 1, alpha * v.y);
}

// ------------------------------------------------------------------ host
extern "C" void kernel_launch(void* const* d_in, const int* in_sizes, int n_in,
                              void* d_out, int out_size, void* d_ws, size_t ws_size,
                              hipStream_t stream) {
    const float* X   = (const float*)d_in[0];
    const float* W   = (const float*)d_in[1];
    const float* att = (const float*)d_in[2];
    const int*   adj = (const int*)  d_in[3];

    const int d_out_dim = in_sizes[2] / 2;            // 64
    const int d_in_dim  = in_sizes[1] / d_out_dim;    // 128
    const int n_nodes   = in_sizes[0] / d_in_dim;     // 100000
    const int n_edges   = in_sizes[3] / 2;            // 1000000
    (void)d_out_dim; (void)d_in_dim;

    const int* esrc = adj;
    const int* etrg = adj + n_edges;

    // workspace layout (floats): wx | s_src | s_trg | denom | segmax | logits
    float*    wx     = (float*)d_ws;
    float*    s_src  = wx + (size_t)n_nodes * D_OUT;
    float*    s_trg  = s_src + n_nodes;
    float*    denom  = s_trg + n_nodes;
    unsigned* segmax = (unsigned*)(denom + n_nodes);
    float*    logits = (float*)(segmax + n_nodes);
    float*    out    = (float*)d_out;

    const int B = 256;

    {   // init: zero out, reset segmax/denom
        int total = n_nodes * D_OUT;
        k_init<<<(total + B - 1) / B, B, 0, stream>>>(out, segmax, denom, n_nodes);
    }
    {   // GEMM: 16-row tile per wave, 8 waves per block
        int tiles  = (n_nodes + 15) / 16;
        int blocks = (tiles + 7) / 8;
        k_gemm<<<blocks, B, 0, stream>>>(X, W, wx, n_nodes);
    }
    k_scores<<<(n_nodes + B - 1) / B, B, 0, stream>>>(wx, att, s_src, s_trg, n_nodes);
    k_logits<<<(n_edges + B - 1) / B, B, 0, stream>>>(esrc, etrg, s_src, s_trg,
                                                      logits, segmax, n_edges);
    k_exp<<<(n_edges + B - 1) / B, B, 0, stream>>>(esrc, logits, segmax, denom, n_edges);
    {   // warp (32 lanes) per edge
        long long threads = (long long)n_edges * 32;
        int blocks = (int)((threads + B - 1) / B);
        k_scatter<<<blocks, B, 0, stream>>>(esrc, etrg, logits, denom, wx, out, n_edges);
    }
}